// UKNFilter_81389630259823
// MI455X (gfx1250) — compile-verified
//
#include <hip/hip_runtime.h>
#include <math.h>

typedef float v2f __attribute__((ext_vector_type(2)));
typedef float v8f __attribute__((ext_vector_type(8)));

#define DTc     0.05f
#define CDAMP   0.25f
#define APc     0.1f
#define AVc     0.1f
#define JITTER  1e-6f
#define FEAT_EPS 1e-8f
#define SQRT_Cc 1.6f
#define DK_SCALE 0.1f

constexpr int SA    = 34;            // LDS row stride in floats (pad vs 64 banks)
constexpr int WAVES = 4;
constexpr int BLOCK = WAVES * 32;

__device__ __forceinline__ float sigmf(float x) { return 1.0f / (1.0f + expf(-x)); }

// ---- fp32 WMMA 16x16x4 helpers ------------------------------------------
__device__ __forceinline__ v8f wmma_acc(v2f a, v2f b, v8f c) {
  return __builtin_amdgcn_wmma_f32_16x16x4_f32(false, a, false, b, (short)0, c,
                                               false, false);
}

// A tile (16 x 4*KCH) from LDS (row stride SA); W in global, natural KxN
// row-major layout, pointer pre-offset to column tile. Layout rule (ISA
// 7.12.2, 32-bit A/B): element k = 4*kk + 2*(lane>>4) + v.
template <int KCH>
__device__ __forceinline__ v8f mm_nat(const float* As, const float* W, int ldn,
                                      int l15, int half) {
  v8f c = {0.f, 0.f, 0.f, 0.f, 0.f, 0.f, 0.f, 0.f};
#pragma unroll
  for (int kk = 0; kk < KCH; ++kk) {
    const int k0 = 4 * kk + 2 * half;
    v2f a = {As[l15 * SA + k0], As[l15 * SA + k0 + 1]};
    v2f b = {W[k0 * ldn + l15], W[(k0 + 1) * ldn + l15]};
    c = wmma_acc(a, b, c);
  }
  return c;
}

// W stored (N,K) row-major (i.e. we need W^T as the B matrix): contiguous in k.
template <int KCH>
__device__ __forceinline__ v8f mm_tr(const float* As, const float* Wt, int ldk,
                                     int l15, int half) {
  v8f c = {0.f, 0.f, 0.f, 0.f, 0.f, 0.f, 0.f, 0.f};
#pragma unroll
  for (int kk = 0; kk < KCH; ++kk) {
    const int k0 = 4 * kk + 2 * half;
    v2f a = {As[l15 * SA + k0], As[l15 * SA + k0 + 1]};
    v2f b = {Wt[l15 * ldk + k0], Wt[l15 * ldk + k0 + 1]};
    c = wmma_acc(a, b, c);
  }
  return c;
}

// Natural-layout W with ncol < 16 valid output columns (zero-pad B).
template <int KCH>
__device__ __forceinline__ v8f mm_nat_masked(const float* As, const float* W,
                                             int ldn, int ncol, int l15,
                                             int half) {
  v8f c = {0.f, 0.f, 0.f, 0.f, 0.f, 0.f, 0.f, 0.f};
  const bool ok = l15 < ncol;
#pragma unroll
  for (int kk = 0; kk < KCH; ++kk) {
    const int k0 = 4 * kk + 2 * half;
    v2f a = {As[l15 * SA + k0], As[l15 * SA + k0 + 1]};
    v2f b = {ok ? W[k0 * ldn + l15] : 0.0f, ok ? W[(k0 + 1) * ldn + l15] : 0.0f};
    c = wmma_acc(a, b, c);
  }
  return c;
}

// C/D layout (ISA 7.12.2): row M = v + 8*(lane>>4), col N = lane&15.
__device__ __forceinline__ void frag_store(float* Os, v8f c, float bias,
                                           bool relu, int l15, int half) {
#pragma unroll
  for (int v = 0; v < 8; ++v) {
    float t = c[v] + bias;
    if (relu) t = fmaxf(t, 0.0f);
    Os[(v + 8 * half) * SA + l15] = t;
  }
}

__global__ __launch_bounds__(BLOCK) void ukf_wmma_kernel(
    const float* __restrict__ zg, const float* __restrict__ Pg,
    const float* __restrict__ ug, const float* __restrict__ yg,
    const float* __restrict__ h0g, const float* __restrict__ zprg,
    const float* __restrict__ Qm, const float* __restrict__ Rm,
    const float* __restrict__ wm, const float* __restrict__ wc,
    const float* __restrict__ ln_g, const float* __restrict__ ln_b,
    const float* __restrict__ w1, const float* __restrict__ b1,
    const float* __restrict__ w2, const float* __restrict__ b2,
    const float* __restrict__ wih, const float* __restrict__ whh,
    const float* __restrict__ bih, const float* __restrict__ bhh,
    const float* __restrict__ wt, const float* __restrict__ bt,
    const float* __restrict__ wd1, const float* __restrict__ bd1,
    const float* __restrict__ wd2, const float* __restrict__ bd2,
    float* __restrict__ out, int B) {
  __shared__ float sA[WAVES][32 * SA];
  __shared__ float sB[WAVES][32 * SA];
  __shared__ float sH[WAVES][32 * SA];

  const int lane = threadIdx.x & 31;
  const int wid = threadIdx.x >> 5;
  const int l15 = lane & 15;
  const int half = lane >> 4;
  const int wbase = (blockIdx.x * WAVES + wid) * 32;
  const int elem = wbase + lane;
  const int eidx = elem < B ? elem : B - 1;   // clamp; full waves keep EXEC=~0

  float* Ab = sA[wid];
  float* Bb = sB[wid];
  float* Hb = sH[wid];

  // ================= Phase 0: per-lane UKF scalar part ====================
  float zv[4], zprev[4], Pm[4][4];
  {
    const float4 t = ((const float4*)zg)[eidx];
    zv[0] = t.x; zv[1] = t.y; zv[2] = t.z; zv[3] = t.w;
    const float4 t2 = ((const float4*)zprg)[eidx];
    zprev[0] = t2.x; zprev[1] = t2.y; zprev[2] = t2.z; zprev[3] = t2.w;
    const float4* p4 = (const float4*)(Pg + (size_t)eidx * 16);
#pragma unroll
    for (int i = 0; i < 4; ++i) {
      float4 r = p4[i];
      Pm[i][0] = r.x; Pm[i][1] = r.y; Pm[i][2] = r.z; Pm[i][3] = r.w;
    }
  }
  const float u = ug[eidx];
  const float2 yv = ((const float2*)yg)[eidx];

  // stage h0 row into Hb
  {
    const float4* hp = (const float4*)(h0g + (size_t)eidx * 32);
#pragma unroll
    for (int i = 0; i < 8; ++i) {
      float4 h4 = hp[i];
      Hb[lane * SA + 4 * i + 0] = h4.x;
      Hb[lane * SA + 4 * i + 1] = h4.y;
      Hb[lane * SA + 4 * i + 2] = h4.z;
      Hb[lane * SA + 4 * i + 3] = h4.w;
    }
  }

  // symmetrize + jitter, then 4x4 Cholesky
  float Ps[4][4];
#pragma unroll
  for (int i = 0; i < 4; ++i)
#pragma unroll
    for (int j = 0; j < 4; ++j) Ps[i][j] = 0.5f * (Pm[i][j] + Pm[j][i]);
#pragma unroll
  for (int i = 0; i < 4; ++i) Ps[i][i] += JITTER;

  float Lm[4][4];
#pragma unroll
  for (int j = 0; j < 4; ++j) {
    float s = Ps[j][j];
#pragma unroll
    for (int k = 0; k < 4; ++k)
      if (k < j) s -= Lm[j][k] * Lm[j][k];
    const float d = sqrtf(s);
    Lm[j][j] = d;
    const float inv = 1.0f / d;
#pragma unroll
    for (int i = 0; i < 4; ++i) {
      if (i > j) {
        float t = Ps[i][j];
#pragma unroll
        for (int k = 0; k < 4; ++k)
          if (k < j) t -= Lm[i][k] * Lm[j][k];
        Lm[i][j] = t * inv;
      } else if (i < j) {
        Lm[i][j] = 0.0f;
      }
    }
  }

  // sigma points through Duffing dynamics
  float X[9][4];
#pragma unroll
  for (int s = 0; s < 9; ++s) {
    float sg[4];
#pragma unroll
    for (int n = 0; n < 4; ++n) {
      float a = 0.0f;
      if (s >= 1 && s <= 4) a = SQRT_Cc * Lm[n][s - 1];
      if (s >= 5) a = -SQRT_Cc * Lm[n][s - 5];
      sg[n] = zv[n] + a;
    }
    const float p = sg[0], v = sg[1], kk = sg[2], al = sg[3];
    X[s][0] = p + DTc * v;
    X[s][1] = v + DTc * (-CDAMP * v - kk * p - al * p * p * p + u);
    X[s][2] = kk;
    X[s][3] = al;
  }

  float wmv[9], wcv[9];
#pragma unroll
  for (int s = 0; s < 9; ++s) { wmv[s] = wm[s]; wcv[s] = wc[s]; }

  float zp[4];
#pragma unroll
  for (int n = 0; n < 4; ++n) {
    float acc = 0.0f;
#pragma unroll
    for (int s = 0; s < 9; ++s) acc += wmv[s] * X[s][n];
    zp[n] = acc;
  }

  // measurement sigma points
  float Y[9][2];
#pragma unroll
  for (int s = 0; s < 9; ++s) {
    const float p = X[s][0], v = X[s][1];
    Y[s][0] = p + APc * p * p * p;
    Y[s][1] = v + AVc * v * v * v;
  }
  float yp[2];
#pragma unroll
  for (int a = 0; a < 2; ++a) {
    float acc = 0.0f;
#pragma unroll
    for (int s = 0; s < 9; ++s) acc += wmv[s] * Y[s][a];
    yp[a] = acc;
  }

  // residual sigma points in place
#pragma unroll
  for (int s = 0; s < 9; ++s) {
#pragma unroll
    for (int n = 0; n < 4; ++n) X[s][n] -= zp[n];
    Y[s][0] -= yp[0];
    Y[s][1] -= yp[1];
  }

  float Pp[4][4];
#pragma unroll
  for (int i = 0; i < 4; ++i)
#pragma unroll
    for (int j = 0; j < 4; ++j) {
      float acc = 0.0f;
#pragma unroll
      for (int s = 0; s < 9; ++s) acc += wcv[s] * X[s][i] * X[s][j];
      Pp[i][j] = acc + Qm[i * 4 + j];
    }
#pragma unroll
  for (int i = 0; i < 4; ++i) Pp[i][i] += JITTER;

  float S1[2][2];
#pragma unroll
  for (int a = 0; a < 2; ++a)
#pragma unroll
    for (int b = 0; b < 2; ++b) {
      float acc = 0.0f;
#pragma unroll
      for (int s = 0; s < 9; ++s) acc += wcv[s] * Y[s][a] * Y[s][b];
      S1[a][b] = acc + Rm[a * 2 + b];
    }
  S1[0][0] += JITTER;
  S1[1][1] += JITTER;

  float Pzy[4][2];
#pragma unroll
  for (int i = 0; i < 4; ++i)
#pragma unroll
    for (int a = 0; a < 2; ++a) {
      float acc = 0.0f;
#pragma unroll
      for (int s = 0; s < 9; ++s) acc += wcv[s] * X[s][i] * Y[s][a];
      Pzy[i][a] = acc;
    }

  // 2x2 Cholesky of sym(S1)+J
  const float c00 = S1[0][0] + JITTER;
  const float c10 = 0.5f * (S1[1][0] + S1[0][1]);
  const float c11 = S1[1][1] + JITTER;
  const float l11 = sqrtf(c00);
  const float l21 = c10 / l11;
  const float l22 = sqrtf(c11 - l21 * l21);
  const float il11 = 1.0f / l11, il22 = 1.0f / l22;

  float Ku[4][2];
#pragma unroll
  for (int i = 0; i < 4; ++i) {
    const float t0 = Pzy[i][0] * il11;
    const float t1 = (Pzy[i][1] - l21 * t0) * il22;
    const float k1 = t1 * il22;
    const float k0 = (t0 - l21 * k1) * il11;
    Ku[i][0] = k0;
    Ku[i][1] = k1;
  }

  const float nu0 = yv.x - yp[0], nu1 = yv.y - yp[1];
  const float tn0 = nu0 * il11;
  const float tn1 = (nu1 - l21 * tn0) * il22;
  const float nis = tn0 * tn0 + tn1 * tn1;
  const float logdet = 2.0f * (logf(l11 + FEAT_EPS) + logf(l22 + FEAT_EPS));

  float feat[20];
  feat[0] = tn0; feat[1] = tn1; feat[2] = nis; feat[3] = logdet;
#pragma unroll
  for (int n = 0; n < 4; ++n) feat[4 + n] = zp[n] - zprev[n];
#pragma unroll
  for (int n = 0; n < 4; ++n) feat[8 + n] = logf(Pp[n][n] + FEAT_EPS);
  feat[12] = Pzy[2][0]; feat[13] = Pzy[2][1];
  feat[14] = Pzy[3][0]; feat[15] = Pzy[3][1];
#pragma unroll
  for (int n = 0; n < 4; ++n) feat[16 + n] = Ku[n][0] * nu0 + Ku[n][1] * nu1;

  float mu = 0.0f;
#pragma unroll
  for (int j = 0; j < 20; ++j) mu += feat[j];
  mu *= (1.0f / 20.0f);
  float var = 0.0f;
#pragma unroll
  for (int j = 0; j < 20; ++j) {
    const float d = feat[j] - mu;
    var += d * d;
  }
  var *= (1.0f / 20.0f);
  const float rinv = 1.0f / sqrtf(var + 1e-5f);
#pragma unroll
  for (int j = 0; j < 20; ++j)
    Ab[lane * SA + j] = (feat[j] - mu) * rinv * ln_g[j] + ln_b[j];

  __syncthreads();

  // ================= MLP via fp32 WMMA ====================================
  // enc1: (32x20) @ (20x32) + b1, relu  -> Bb
#pragma unroll
  for (int mt = 0; mt < 2; ++mt)
#pragma unroll
    for (int nt = 0; nt < 2; ++nt) {
      v8f c = mm_nat<5>(Ab + mt * 16 * SA, w1 + nt * 16, 32, l15, half);
      frag_store(Bb + mt * 16 * SA + nt * 16, c, b1[nt * 16 + l15], true, l15, half);
    }
  __syncthreads();

  // enc2: (32x32) @ (32x32) + b2, relu  -> Ab
#pragma unroll
  for (int mt = 0; mt < 2; ++mt)
#pragma unroll
    for (int nt = 0; nt < 2; ++nt) {
      v8f c = mm_nat<8>(Bb + mt * 16 * SA, w2 + nt * 16, 32, l15, half);
      frag_store(Ab + mt * 16 * SA + nt * 16, c, b2[nt * 16 + l15], true, l15, half);
    }
  __syncthreads();

  // GRU: gi = X2 @ wih^T + bih ; gh = h0 @ whh^T + bhh ; fuse elementwise.
  for (int mt = 0; mt < 2; ++mt) {
    v8f gi[6], gh[6];
    const float* As = Ab + mt * 16 * SA;
    const float* Hs = Hb + mt * 16 * SA;
#pragma unroll
    for (int nt = 0; nt < 6; ++nt) {
      v8f c = mm_tr<8>(As, wih + nt * 16 * 32, 32, l15, half);
      const float bi = bih[nt * 16 + l15];
#pragma unroll
      for (int v = 0; v < 8; ++v) c[v] += bi;
      gi[nt] = c;
      v8f d = mm_tr<8>(Hs, whh + nt * 16 * 32, 32, l15, half);
      const float bh = bhh[nt * 16 + l15];
#pragma unroll
      for (int v = 0; v < 8; ++v) d[v] += bh;
      gh[nt] = d;
    }
#pragma unroll
    for (int p = 0; p < 2; ++p)
#pragma unroll
      for (int v = 0; v < 8; ++v) {
        const float r = sigmf(gi[p][v] + gh[p][v]);
        const float zgt = sigmf(gi[2 + p][v] + gh[2 + p][v]);
        const float ng = tanhf(gi[4 + p][v] + r * gh[4 + p][v]);
        const int off = (mt * 16 + v + 8 * half) * SA + p * 16 + l15;
        Bb[off] = (1.0f - zgt) * ng + zgt * Hb[off];
      }
  }
  __syncthreads();

  // trunk: h_new @ trunk_w + bt, relu -> Ab
#pragma unroll
  for (int mt = 0; mt < 2; ++mt)
#pragma unroll
    for (int nt = 0; nt < 2; ++nt) {
      v8f c = mm_nat<8>(Bb + mt * 16 * SA, wt + nt * 16, 32, l15, half);
      frag_store(Ab + mt * 16 * SA + nt * 16, c, bt[nt * 16 + l15], true, l15, half);
    }
  __syncthreads();

  // dk hidden: t @ dk_w1 + bd1, relu -> Bb
#pragma unroll
  for (int mt = 0; mt < 2; ++mt)
#pragma unroll
    for (int nt = 0; nt < 2; ++nt) {
      v8f c = mm_nat<8>(Ab + mt * 16 * SA, wd1 + nt * 16, 32, l15, half);
      frag_store(Bb + mt * 16 * SA + nt * 16, c, bd1[nt * 16 + l15], true, l15, half);
    }
  __syncthreads();

  // dk out: (32x32) @ (32x8) + bd2 (no relu) -> Hb cols 0..7
#pragma unroll
  for (int mt = 0; mt < 2; ++mt) {
    v8f c = mm_nat_masked<8>(Bb + mt * 16 * SA, wd2, 8, 8, l15, half);
    if (l15 < 8) {
      const float bb = bd2[l15];
#pragma unroll
      for (int v = 0; v < 8; ++v)
        Hb[(mt * 16 + v + 8 * half) * SA + l15] = c[v] + bb;
    }
  }
  __syncthreads();

  // ================= Finale: Kalman update per lane =======================
  float dkraw[8];
#pragma unroll
  for (int j = 0; j < 8; ++j) dkraw[j] = Hb[lane * SA + j];

  float nrm2 = 0.0f;
#pragma unroll
  for (int i = 0; i < 4; ++i) nrm2 += Ku[i][0] * Ku[i][0] + Ku[i][1] * Ku[i][1];
  const float scale = DK_SCALE * sqrtf(nrm2);

  float Kf[4][2];
#pragma unroll
  for (int i = 0; i < 4; ++i)
#pragma unroll
    for (int a = 0; a < 2; ++a)
      Kf[i][a] = Ku[i][a] + scale * tanhf(dkraw[i * 2 + a]);

  float zpost[4];
#pragma unroll
  for (int i = 0; i < 4; ++i) zpost[i] = zp[i] + Kf[i][0] * nu0 + Kf[i][1] * nu1;

  float KS[4][2];
#pragma unroll
  for (int i = 0; i < 4; ++i)
#pragma unroll
    for (int a = 0; a < 2; ++a)
      KS[i][a] = Kf[i][0] * S1[0][a] + Kf[i][1] * S1[1][a];

  float Ppost[4][4];
#pragma unroll
  for (int i = 0; i < 4; ++i)
#pragma unroll
    for (int j = 0; j < 4; ++j) {
      const float m1 = Kf[i][0] * Pzy[j][0] + Kf[i][1] * Pzy[j][1];
      const float m2 = Pzy[i][0] * Kf[j][0] + Pzy[i][1] * Kf[j][1];
      const float m3 = KS[i][0] * Kf[j][0] + KS[i][1] * Kf[j][1];
      Ppost[i][j] = Pp[i][j] - m1 - m2 + m3;
    }
#pragma unroll
  for (int i = 0; i < 4; ++i)
#pragma unroll
    for (int j = 0; j < 4; ++j)
      if (j > i) {
        const float av = 0.5f * (Ppost[i][j] + Ppost[j][i]);
        Ppost[i][j] = av;
        Ppost[j][i] = av;
      }
#pragma unroll
  for (int i = 0; i < 4; ++i) Ppost[i][i] += JITTER;

  // stage 20-wide rows in LDS, then coalesced store
  float* Ob = Ab;  // dead buffer, 1088 floats >= 32*20
#pragma unroll
  for (int i = 0; i < 4; ++i) Ob[lane * 20 + i] = zpost[i];
#pragma unroll
  for (int i = 0; i < 4; ++i)
#pragma unroll
    for (int j = 0; j < 4; ++j) Ob[lane * 20 + 4 + i * 4 + j] = Ppost[i][j];
  __syncthreads();

  const size_t obase = (size_t)wbase * 20;
  const size_t olim = (size_t)B * 20;
#pragma unroll
  for (int i = 0; i < 20; ++i) {
    const size_t oidx = obase + (size_t)i * 32 + lane;
    if (oidx < olim) out[oidx] = Ob[i * 32 + lane];
  }
}

extern "C" void kernel_launch(void* const* d_in, const int* in_sizes, int n_in,
                              void* d_out, int out_size, void* d_ws,
                              size_t ws_size, hipStream_t stream) {
  const float* zg  = (const float*)d_in[0];
  const float* Pg  = (const float*)d_in[1];
  const float* ug  = (const float*)d_in[2];
  const float* yg  = (const float*)d_in[3];
  const float* h0g = (const float*)d_in[4];
  const float* zpr = (const float*)d_in[5];
  const float* Qm  = (const float*)d_in[6];
  const float* Rm  = (const float*)d_in[7];
  const float* wm  = (const float*)d_in[8];
  const float* wc  = (const float*)d_in[9];
  const float* lng = (const float*)d_in[10];
  const float* lnb = (const float*)d_in[11];
  const float* w1  = (const float*)d_in[12];
  const float* b1  = (const float*)d_in[13];
  const float* w2  = (const float*)d_in[14];
  const float* b2  = (const float*)d_in[15];
  const float* wih = (const float*)d_in[16];
  const float* whh = (const float*)d_in[17];
  const float* bih = (const float*)d_in[18];
  const float* bhh = (const float*)d_in[19];
  const float* wt  = (const float*)d_in[20];
  const float* bt  = (const float*)d_in[21];
  const float* wd1 = (const float*)d_in[22];
  const float* bd1 = (const float*)d_in[23];
  const float* wd2 = (const float*)d_in[24];
  const float* bd2 = (const float*)d_in[25];
  // d_in[26], d_in[27] (gate_w, gate_b) feed only rho_x/rho_theta -> dead.

  const int B = in_sizes[0] / 4;
  const int blocks = (B + BLOCK - 1) / BLOCK;
  ukf_wmma_kernel<<<blocks, BLOCK, 0, stream>>>(
      zg, Pg, ug, yg, h0g, zpr, Qm, Rm, wm, wc, lng, lnb, w1, b1, w2, b2, wih,
      whh, bih, bhh, wt, bt, wd1, bd1, wd2, bd2, (float*)d_out, B);
}